// CombTwoNodes_CYK_71837622992941
// MI455X (gfx1250) — compile-verified
//
#include <hip/hip_runtime.h>
#include <math.h>

typedef __attribute__((ext_vector_type(2))) float v2f;
typedef __attribute__((ext_vector_type(8))) float v8f;

#define DDIM   256
#define OUTH   512
#define CANDI  256
#define NODES  256
#define LTOT   (NODES * CANDI)
#define NPOS   128

// B LDS layout: k-pair interleaved. Pair row kp holds (2kp+p, n) at [kp*PRB + 2n + p].
// PRB = 1024 + 32 pad: row delta ≡ 32 (mod 64 banks) -> halves' banks disjoint.
#define PRB 1056

// workspace layout (float offsets)
#define WS_SCORE   0              // 65536 floats
#define WS_MAXVAL  65536          // 256 floats
#define WS_MAXIX   65792          // 256 ints
#define WS_FEAT1   66048          // 256*512 = 131072
#define WS_LC      197120         // 65536
#define WS_RC      262656         // 65536
#define WS_GATES   328192         // 5*256*512 = 655360

static __device__ __forceinline__ v8f wmma4(v2f a, v2f b, v8f c) {
  // D = A(16x4 f32) * B(4x16 f32) + C(16x16 f32), wave32 striped
  return __builtin_amdgcn_wmma_f32_16x16x4_f32(false, a, false, b, (short)0, c,
                                               false, false);
}

static __device__ __forceinline__ float sigmoidf_(float x) {
  return 1.0f / (1.0f + __expf(-x));
}

// ---------------------------------------------------------------------------
// Kernel 1: score2 = sigmoid(tanh(feat2 @ W_cst + b_cst) @ W_out + b_out)
// feat2 gathered on the fly from input_h via 4 index arrays.
// One WG: 32 rows x 512 cols, K=1024 tiled by 16 through LDS.
// ---------------------------------------------------------------------------
__global__ __launch_bounds__(256) void k_score_gemm(
    const float* __restrict__ input_h,
    const int* __restrict__ ixL, const int* __restrict__ ixR,
    const int* __restrict__ ixSL, const int* __restrict__ ixSR,
    const float* __restrict__ Wc, const float* __restrict__ bc,
    const float* __restrict__ Wout, const float* __restrict__ bout,
    float* __restrict__ score)
{
  __shared__ float A[32 * 20];    // 32 rows x 16 k, stride 20 (pad)
  __shared__ float B[8 * PRB];    // 8 k-pairs x 512 n, interleaved pairs
  __shared__ float S[32];         // per-row partial score

  const int tid  = threadIdx.x;
  const int wave = tid >> 5;
  const int lane = tid & 31;
  const int half = lane >> 4;
  const int lm   = lane & 15;
  const int rowBase = blockIdx.x * 32;

  if (tid < 32) S[tid] = 0.0f;

  v8f acc[2][4];
#pragma unroll
  for (int mt = 0; mt < 2; ++mt)
#pragma unroll
    for (int nt = 0; nt < 4; ++nt)
#pragma unroll
      for (int e = 0; e < 8; ++e) acc[mt][nt][e] = 0.0f;

  for (int ch = 0; ch < 64; ++ch) {
    const int seg    = ch >> 4;          // which of the 4 gathered blocks
    const int coloff = (ch & 15) * 16;   // column offset inside that block
    const int kbase  = ch * 16;
    const int* idxArr = (seg == 0) ? ixL : (seg == 1) ? ixR
                       : (seg == 2) ? ixSL : ixSR;
    __syncthreads();
    // stage A: 32 rows x 16 cols (gathered rows of input_h)
    if (tid < 128) {
      const int row = tid >> 2, c4 = tid & 3;
      const int ix = idxArr[rowBase + row];
      const float4 v =
          *(const float4*)&input_h[(size_t)ix * DDIM + coloff + c4 * 4];
      *(float4*)&A[row * 20 + c4 * 4] = v;
    }
    // stage B: 16 rows x 512 cols of W_cst, interleaved into k-pairs
    for (int i = tid; i < 8 * 128; i += 256) {
      const int kp = i >> 7, c4 = i & 127;
      const int gk = kbase + 2 * kp;
      const float4 x = *(const float4*)&Wc[(size_t)gk * OUTH + c4 * 4];
      const float4 y = *(const float4*)&Wc[(size_t)(gk + 1) * OUTH + c4 * 4];
      float4 lo, hi;
      lo.x = x.x; lo.y = y.x; lo.z = x.y; lo.w = y.y;
      hi.x = x.z; hi.y = y.z; hi.z = x.w; hi.w = y.w;
      *(float4*)&B[kp * PRB + c4 * 8]     = lo;
      *(float4*)&B[kp * PRB + c4 * 8 + 4] = hi;
      if (ch + 1 < 64) {
        __builtin_prefetch(&Wc[(size_t)(gk + 16) * OUTH + c4 * 4], 0, 1);
        __builtin_prefetch(&Wc[(size_t)(gk + 17) * OUTH + c4 * 4], 0, 1);
      }
    }
    __syncthreads();

#pragma unroll
    for (int ks = 0; ks < 4; ++ks) {
      const int kk = ks * 4 + 2 * half;   // even k of this lane-half's pair
      const int kp = ks * 2 + half;       // pair row in LDS
      const v2f a0 = *(const v2f*)&A[(0 * 16 + lm) * 20 + kk];
      const v2f a1 = *(const v2f*)&A[(1 * 16 + lm) * 20 + kk];
#pragma unroll
      for (int nt = 0; nt < 4; ++nt) {
        const int col = wave * 64 + nt * 16 + lm;
        const v2f b = *(const v2f*)&B[kp * PRB + col * 2];
        acc[0][nt] = wmma4(a0, b, acc[0][nt]);
        acc[1][nt] = wmma4(a1, b, acc[1][nt]);
      }
    }
  }
  __syncthreads();

  // epilogue: tanh(+bias) dot W_out, reduce per row in LDS
#pragma unroll
  for (int mt = 0; mt < 2; ++mt) {
#pragma unroll
    for (int r = 0; r < 8; ++r) {
      float s = 0.0f;
#pragma unroll
      for (int nt = 0; nt < 4; ++nt) {
        const int col = wave * 64 + nt * 16 + lm;
        const float h = tanhf(acc[mt][nt][r] + bc[col]);
        s += h * Wout[col];
      }
      atomicAdd(&S[mt * 16 + r + 8 * half], s);
    }
  }
  __syncthreads();
  if (tid < 32) {
    const float t = S[tid] + bout[0];
    score[rowBase + tid] = sigmoidf_(t);
  }
}

// ---------------------------------------------------------------------------
// Kernel 2: per-node max / argmax over candi (first-max tie-break)
// ---------------------------------------------------------------------------
__global__ __launch_bounds__(256) void k_argmax(
    const float* __restrict__ score, float* __restrict__ maxval,
    int* __restrict__ maxix)
{
  __shared__ float v[256];
  __shared__ int ix[256];
  const int n = blockIdx.x, t = threadIdx.x;
  v[t] = score[n * CANDI + t];
  ix[t] = t;
  __syncthreads();
  for (int s = 128; s > 0; s >>= 1) {
    if (t < s) {
      if (v[t + s] > v[t]) { v[t] = v[t + s]; ix[t] = ix[t + s]; }
    }
    __syncthreads();
  }
  if (t == 0) { maxval[n] = v[0]; maxix[n] = ix[0]; }
}

// ---------------------------------------------------------------------------
// Kernel 3: gather only the argmax-selected rows -> feat1_sel, lc, rc
// ---------------------------------------------------------------------------
__global__ __launch_bounds__(256) void k_gather_sel(
    const float* __restrict__ input_h, const float* __restrict__ input_c,
    const int* __restrict__ ixL, const int* __restrict__ ixR,
    const int* __restrict__ maxix,
    float* __restrict__ feat1, float* __restrict__ lc, float* __restrict__ rc)
{
  const int n = blockIdx.x, t = threadIdx.x;
  const int j = n * CANDI + maxix[n];
  const int part = t >> 6;
  const int c4 = (t & 63) * 4;
  const float* src;
  float* dst;
  if (part == 0)      { src = &input_h[(size_t)ixL[j] * DDIM]; dst = &feat1[n * 512]; }
  else if (part == 1) { src = &input_h[(size_t)ixR[j] * DDIM]; dst = &feat1[n * 512 + 256]; }
  else if (part == 2) { src = &input_c[(size_t)ixL[j] * DDIM]; dst = &lc[n * 256]; }
  else                { src = &input_c[(size_t)ixR[j] * DDIM]; dst = &rc[n * 256]; }
  *(float4*)&dst[c4] = *(const float4*)&src[c4];
}

// ---------------------------------------------------------------------------
// Kernel 4: 5 gate GEMMs [256,512] x [512,512] with fused activation
// blockIdx.x = row block (32 rows), blockIdx.y = gate index
// ---------------------------------------------------------------------------
__global__ __launch_bounds__(256) void k_gates_gemm(
    const float* __restrict__ feat1,
    const float* __restrict__ Wu, const float* __restrict__ bu,
    const float* __restrict__ Wi, const float* __restrict__ bi,
    const float* __restrict__ Wfl, const float* __restrict__ bfl,
    const float* __restrict__ Wfr, const float* __restrict__ bfr,
    const float* __restrict__ Wo, const float* __restrict__ bo,
    float* __restrict__ gates)
{
  __shared__ float A[32 * 20];
  __shared__ float B[8 * PRB];
  const int tid = threadIdx.x, wave = tid >> 5, lane = tid & 31;
  const int half = lane >> 4, lm = lane & 15;
  const int rowBase = blockIdx.x * 32;
  const int g = blockIdx.y;
  const float *W, *bias;
  switch (g) {
    case 0:  W = Wu;  bias = bu;  break;
    case 1:  W = Wi;  bias = bi;  break;
    case 2:  W = Wfl; bias = bfl; break;
    case 3:  W = Wfr; bias = bfr; break;
    default: W = Wo;  bias = bo;  break;
  }

  v8f acc[2][4];
#pragma unroll
  for (int mt = 0; mt < 2; ++mt)
#pragma unroll
    for (int nt = 0; nt < 4; ++nt)
#pragma unroll
      for (int e = 0; e < 8; ++e) acc[mt][nt][e] = 0.0f;

  for (int ch = 0; ch < 32; ++ch) {
    const int kbase = ch * 16;
    __syncthreads();
    if (tid < 128) {
      const int row = tid >> 2, c4 = tid & 3;
      *(float4*)&A[row * 20 + c4 * 4] =
          *(const float4*)&feat1[(size_t)(rowBase + row) * 512 + kbase + c4 * 4];
    }
    for (int i = tid; i < 8 * 128; i += 256) {
      const int kp = i >> 7, c4 = i & 127;
      const int gk = kbase + 2 * kp;
      const float4 x = *(const float4*)&W[(size_t)gk * 512 + c4 * 4];
      const float4 y = *(const float4*)&W[(size_t)(gk + 1) * 512 + c4 * 4];
      float4 lo, hi;
      lo.x = x.x; lo.y = y.x; lo.z = x.y; lo.w = y.y;
      hi.x = x.z; hi.y = y.z; hi.z = x.w; hi.w = y.w;
      *(float4*)&B[kp * PRB + c4 * 8]     = lo;
      *(float4*)&B[kp * PRB + c4 * 8 + 4] = hi;
    }
    __syncthreads();
#pragma unroll
    for (int ks = 0; ks < 4; ++ks) {
      const int kk = ks * 4 + 2 * half;
      const int kp = ks * 2 + half;
      const v2f a0 = *(const v2f*)&A[(0 * 16 + lm) * 20 + kk];
      const v2f a1 = *(const v2f*)&A[(1 * 16 + lm) * 20 + kk];
#pragma unroll
      for (int nt = 0; nt < 4; ++nt) {
        const int col = wave * 64 + nt * 16 + lm;
        const v2f b = *(const v2f*)&B[kp * PRB + col * 2];
        acc[0][nt] = wmma4(a0, b, acc[0][nt]);
        acc[1][nt] = wmma4(a1, b, acc[1][nt]);
      }
    }
  }

  float* out = &gates[(size_t)g * NODES * 512];
#pragma unroll
  for (int mt = 0; mt < 2; ++mt)
#pragma unroll
    for (int r = 0; r < 8; ++r) {
      const int row = rowBase + mt * 16 + r + 8 * half;
#pragma unroll
      for (int nt = 0; nt < 4; ++nt) {
        const int col = wave * 64 + nt * 16 + lm;
        float v = acc[mt][nt][r] + bias[col];
        v = (g == 0) ? tanhf(v) : sigmoidf_(v);
        out[(size_t)row * 512 + col] = v;
      }
    }
}

// ---------------------------------------------------------------------------
// Kernel 5: LSTM cell combine + head gating -> h_1, c_1
// ---------------------------------------------------------------------------
__global__ __launch_bounds__(256) void k_combine(
    const float* __restrict__ gates, const float* __restrict__ lc,
    const float* __restrict__ rc, const float* __restrict__ maxval,
    float* __restrict__ out_h1, float* __restrict__ out_c1)
{
  const int n = blockIdx.x, d = threadIdx.x;
  const float lcv = lc[n * 256 + d], rcv = rc[n * 256 + d];
  const float w0 = maxval[n], w1 = 1.0f - w0;
  float csum = 0.0f, hsum = 0.0f;
#pragma unroll
  for (int h = 0; h < 2; ++h) {
    const size_t base = (size_t)n * 512 + h * 256 + d;
    const float u  = gates[(size_t)0 * NODES * 512 + base];
    const float ig = gates[(size_t)1 * NODES * 512 + base];
    const float fl = gates[(size_t)2 * NODES * 512 + base];
    const float fr = gates[(size_t)3 * NODES * 512 + base];
    const float og = gates[(size_t)4 * NODES * 512 + base];
    const float c1 = fl * lcv + fr * rcv + ig * u;
    const float h1 = og * tanhf(c1);
    const float w = (h == 0) ? w0 : w1;
    csum += c1 * w;
    hsum += h1 * w;
  }
  out_h1[n * 256 + d] = hsum;
  out_c1[n * 256 + d] = csum;
}

// ---------------------------------------------------------------------------
// Kernel 6: masked selections + gate losses
// ---------------------------------------------------------------------------
__global__ __launch_bounds__(256) void k_losses(
    const float* __restrict__ score, const float* __restrict__ maxval,
    const int* __restrict__ P, const int* __restrict__ N,
    const int* __restrict__ G,
    float* __restrict__ gold_p, float* __restrict__ gold_n,
    float* __restrict__ gate_loss, float* __restrict__ gate_a)
{
  __shared__ int ixp[NPOS];
  __shared__ int ixn[NPOS];
  const int t = threadIdx.x;
  if (t == 0) {
    int cp = 0, cn = 0;
    for (int i = 0; i < NODES; ++i) {
      if (P[i] != 0 && cp < NPOS) ixp[cp++] = i;
      if (N[i] != 0 && cn < NPOS) ixn[cn++] = i;
    }
  }
  __syncthreads();
  if (t < NPOS) {
    gold_p[t] = maxval[ixp[t]];
    gold_n[t] = maxval[ixn[t]];
  }
  for (int e = t; e < NPOS * CANDI; e += 256) {
    const int i = e >> 8, k = e & 255;
    const int row = ixp[i];
    const float gold = score[row * CANDI + G[row]];
    const float diff = score[row * CANDI + k] - gold;
    gate_loss[e] = diff > 0.0f ? diff : 0.0f;
  }
  if (t < NPOS) {
    const int row = ixp[t];
    const float gold = score[row * CANDI + G[row]];
    float m = 0.0f;
    for (int k = 0; k < CANDI; ++k) {
      const float diff = score[row * CANDI + k] - gold;
      if (diff > m) m = diff;
    }
    gate_a[t] = m;
  }
}

// ---------------------------------------------------------------------------
extern "C" void kernel_launch(void* const* d_in, const int* in_sizes, int n_in,
                              void* d_out, int out_size, void* d_ws,
                              size_t ws_size, hipStream_t stream)
{
  const float* input_h = (const float*)d_in[0];
  const float* input_c = (const float*)d_in[1];
  const int* ixL  = (const int*)d_in[2];
  const int* ixR  = (const int*)d_in[3];
  const int* ixSL = (const int*)d_in[4];
  const int* ixSR = (const int*)d_in[5];
  const int* P    = (const int*)d_in[6];
  const int* Nm   = (const int*)d_in[7];
  const int* G    = (const int*)d_in[8];
  const float* Wc   = (const float*)d_in[13];
  const float* bc   = (const float*)d_in[14];
  const float* Wout = (const float*)d_in[15];
  const float* bout = (const float*)d_in[16];
  const float* Wu  = (const float*)d_in[17]; const float* bu  = (const float*)d_in[18];
  const float* Wi  = (const float*)d_in[19]; const float* bi  = (const float*)d_in[20];
  const float* Wfl = (const float*)d_in[21]; const float* bfl = (const float*)d_in[22];
  const float* Wfr = (const float*)d_in[23]; const float* bfr = (const float*)d_in[24];
  const float* Wo  = (const float*)d_in[25]; const float* bo  = (const float*)d_in[26];

  float* ws     = (float*)d_ws;
  float* score  = ws + WS_SCORE;
  float* maxval = ws + WS_MAXVAL;
  int*   maxix  = (int*)(ws + WS_MAXIX);
  float* feat1  = ws + WS_FEAT1;
  float* lc     = ws + WS_LC;
  float* rc     = ws + WS_RC;
  float* gates  = ws + WS_GATES;

  float* out       = (float*)d_out;
  float* out_h1    = out;                 // 65536
  float* out_c1    = out + 65536;         // 65536
  float* gold_p    = out + 131072;        // 128
  float* gold_n    = out + 131200;        // 128
  float* gate_loss = out + 131328;        // 32768
  float* gate_a    = out + 164096;        // 128

  k_score_gemm<<<LTOT / 32, 256, 0, stream>>>(input_h, ixL, ixR, ixSL, ixSR,
                                              Wc, bc, Wout, bout, score);
  k_argmax<<<NODES, 256, 0, stream>>>(score, maxval, maxix);
  k_gather_sel<<<NODES, 256, 0, stream>>>(input_h, input_c, ixL, ixR, maxix,
                                          feat1, lc, rc);
  k_gates_gemm<<<dim3(NODES / 32, 5), 256, 0, stream>>>(
      feat1, Wu, bu, Wi, bi, Wfl, bfl, Wfr, bfr, Wo, bo, gates);
  k_combine<<<NODES, 256, 0, stream>>>(gates, lc, rc, maxval, out_h1, out_c1);
  k_losses<<<1, 256, 0, stream>>>(score, maxval, P, Nm, G, gold_p, gold_n,
                                  gate_loss, gate_a);
}